// VectorQuantizer_20298015441328
// MI455X (gfx1250) — compile-verified
//
#include <hip/hip_runtime.h>
#include <hip/hip_bf16.h>

// ---- problem constants ---------------------------------------------------
#define DIMC   64          // channels / vector dim
#define HW     4096        // 64*64 spatial
#define NCODE  512         // codebook entries
#define NROWS  131072      // 32*64*64 flattened vectors
#define NELEM  8388608.f   // NROWS * DIMC
#define NBLK   2048        // main-kernel grid (NROWS / 64 rows per block)

typedef float v2f __attribute__((ext_vector_type(2)));
typedef float v4f __attribute__((ext_vector_type(4)));
typedef float v8f __attribute__((ext_vector_type(8)));

// ---- kernel 1: codebook half-norms + zero scratch ------------------------
__global__ void vq_prep(const float* __restrict__ cb,
                        float* __restrict__ hnorm,
                        unsigned* __restrict__ counts,
                        float* __restrict__ partials) {
    const int tid = blockIdx.x * blockDim.x + threadIdx.x;
    if (tid < NCODE) {
        float s = 0.f;
        #pragma unroll
        for (int d = 0; d < DIMC; ++d) {
            float v = cb[tid * DIMC + d];
            s += v * v;
        }
        hnorm[tid]  = 0.5f * s;
        counts[tid] = 0u;
    }
    if (tid < NBLK) partials[tid] = 0.f;
}

// ---- kernel 2: distances (WMMA f32), argmax, gather, one-hot -------------
__global__ __launch_bounds__(128) void vq_main(
        const float* __restrict__ x,        // NCHW (32,64,64,64)
        const float* __restrict__ cb,       // (512,64)
        const float* __restrict__ hnorm,    // (512)
        unsigned*    __restrict__ counts,   // (512)
        float*       __restrict__ partials, // (NBLK)
        float*       __restrict__ qout,     // NCHW quantized
        float*       __restrict__ enc)      // (NROWS, 512) one-hot
{
    __shared__ int   lds_rows[4][16];
    __shared__ float lds_loss[4];

    const int lane = threadIdx.x & 31;
    const int wave = threadIdx.x >> 5;
    const int gw   = blockIdx.x * 4 + wave;   // global wave id
    const int n0   = gw * 16;                 // first of this wave's 16 rows
    const int b    = n0 >> 12;                // batch  (HW = 4096)
    const int p0   = n0 & 4095;               // spatial offset (16-aligned)
    const size_t xbase = (size_t)b * (DIMC * HW) + (size_t)p0;

    // warm L2 with the codebook (gfx1250 global_prefetch_b8)
    if (wave == 0 && (blockIdx.x & 63) == 0) {
        #pragma unroll 4
        for (int i = lane; i < 1024; i += 32)
            __builtin_prefetch(cb + (size_t)i * 32, 0, 1);
    }

    // --- A fragments: 16 rows x 64 dims of x, laid out for 16x16x4 f32 WMMA
    // lane 0-15: M=lane, K = 4*ks + {0,1}; lane 16-31: M=lane-16, K = 4*ks + {2,3}
    const int m  = lane & 15;
    const int kh = (lane >> 4) * 2;
    v2f av[16];
    #pragma unroll
    for (int ks = 0; ks < 16; ++ks) {
        const int k = ks * 4 + kh;
        av[ks].x = x[xbase + (size_t)k       * HW + m];
        av[ks].y = x[xbase + (size_t)(k + 1) * HW + m];
    }

    // --- score all 512 codes: 32 tiles of 16 codes, K accumulated by WMMA
    float best[8];
    int   bcode[8];
    #pragma unroll
    for (int r = 0; r < 8; ++r) { best[r] = -3.4e38f; bcode[r] = 0; }

    const int col = lane & 15;
    for (int t = 0; t < 32; ++t) {
        const int   code = t * 16 + col;          // this lane's column code
        const float hn   = hnorm[code];
        const float* bp  = cb + (size_t)code * DIMC + kh;

        v8f acc = {0.f, 0.f, 0.f, 0.f, 0.f, 0.f, 0.f, 0.f};
        #pragma unroll
        for (int ks = 0; ks < 16; ++ks) {
            v2f bf = *(const v2f*)(bp + ks * 4);  // codebook[code][4ks+kh .. +1]
            acc = __builtin_amdgcn_wmma_f32_16x16x4_f32(
                    false, av[ks], false, bf, (short)0, acc, false, false);
        }
        // acc[r]: lanes 0-15 -> row r, lanes 16-31 -> row r+8 ; col = lane&15
        #pragma unroll
        for (int r = 0; r < 8; ++r) {
            float s = acc[r] - hn;                // x.c - 0.5||c||^2
            if (s > best[r]) { best[r] = s; bcode[r] = code; }
        }
    }

    // --- argmax across the 16 columns of each half-wave (wave32 shfl_xor)
    #pragma unroll
    for (int r = 0; r < 8; ++r) {
        #pragma unroll
        for (int off = 1; off < 16; off <<= 1) {
            float vs = __shfl_xor(best[r],  off, 32);
            int   vi = __shfl_xor(bcode[r], off, 32);
            if (vs > best[r] || (vs == best[r] && vi < bcode[r])) {
                best[r] = vs; bcode[r] = vi;
            }
        }
        if ((lane & 15) == 0) {                   // lane 0 -> row r, lane 16 -> row r+8
            lds_rows[wave][r + (lane >> 4) * 8] = bcode[r];
            atomicAdd(&counts[bcode[r]], 1u);     // integer atomics: deterministic
        }
    }
    __syncthreads();

    // --- quantized (gather codebook rows) back to NCHW + SSE, coalesced on W
    float lsum = 0.f;
    const int part = lane & 3;                    // which 4-row group
    const int r0   = part * 4;
    const int i0 = lds_rows[wave][r0 + 0];
    const int i1 = lds_rows[wave][r0 + 1];
    const int i2 = lds_rows[wave][r0 + 2];
    const int i3 = lds_rows[wave][r0 + 3];
    #pragma unroll
    for (int pass = 0; pass < 8; ++pass) {
        const int c = pass * 8 + (lane >> 2);     // channel
        v4f q;
        q.x = cb[(size_t)i0 * DIMC + c];
        q.y = cb[(size_t)i1 * DIMC + c];
        q.z = cb[(size_t)i2 * DIMC + c];
        q.w = cb[(size_t)i3 * DIMC + c];
        const size_t off = xbase + (size_t)c * HW + r0;
        v4f xin = *(const v4f*)(x + off);
        v4f d = q - xin;
        lsum += d.x * d.x + d.y * d.y + d.z * d.z + d.w * d.w;
        *(v4f*)(qout + off) = q;
    }

    // --- one-hot encodings: each row = 2KB contiguous, b128 stores
    const size_t ebase = (size_t)n0 * NCODE + (size_t)lane * 16;
    const int bc = lane * 16;
    #pragma unroll 4
    for (int j = 0; j < 16; ++j) {
        const int rj = lds_rows[wave][j];
        float* dst = enc + ebase + (size_t)j * NCODE;
        #pragma unroll
        for (int q4 = 0; q4 < 4; ++q4) {
            v4f v;
            v.x = (bc + q4 * 4 + 0 == rj) ? 1.f : 0.f;
            v.y = (bc + q4 * 4 + 1 == rj) ? 1.f : 0.f;
            v.z = (bc + q4 * 4 + 2 == rj) ? 1.f : 0.f;
            v.w = (bc + q4 * 4 + 3 == rj) ? 1.f : 0.f;
            *(v4f*)(dst + q4 * 4) = v;
        }
    }

    // --- per-block SSE partial (deterministic; no float atomics)
    #pragma unroll
    for (int off = 16; off >= 1; off >>= 1) lsum += __shfl_xor(lsum, off, 32);
    if (lane == 0) lds_loss[wave] = lsum;
    __syncthreads();
    if (threadIdx.x == 0)
        partials[blockIdx.x] = (lds_loss[0] + lds_loss[1]) + (lds_loss[2] + lds_loss[3]);
}

// ---- kernel 3: loss + perplexity (fixed-order reductions) ----------------
__global__ __launch_bounds__(256) void vq_final(
        const unsigned* __restrict__ counts,
        const float*    __restrict__ partials,
        float* __restrict__ out_loss,
        float* __restrict__ out_perp)
{
    __shared__ float sred[256];
    const int tid = threadIdx.x;

    float s = 0.f;
    for (int i = tid; i < NBLK; i += 256) s += partials[i];
    sred[tid] = s;
    __syncthreads();
    for (int st = 128; st > 0; st >>= 1) {
        if (tid < st) sred[tid] += sred[tid + st];
        __syncthreads();
    }
    float loss = 0.f;
    if (tid == 0) loss = sred[0] * (1.25f / NELEM);   // q_loss + 0.25*e_loss
    __syncthreads();

    float e = 0.f;
    for (int k = tid; k < NCODE; k += 256) {
        float p = (float)counts[k] * (1.0f / (float)NROWS);
        e += p * logf(p + 1e-10f);
    }
    sred[tid] = e;
    __syncthreads();
    for (int st = 128; st > 0; st >>= 1) {
        if (tid < st) sred[tid] += sred[tid + st];
        __syncthreads();
    }
    if (tid == 0) {
        *out_loss = loss;
        *out_perp = expf(-sred[0]);
    }
}

// ---- host-side launch ----------------------------------------------------
extern "C" void kernel_launch(void* const* d_in, const int* in_sizes, int n_in,
                              void* d_out, int out_size, void* d_ws, size_t ws_size,
                              hipStream_t stream) {
    const float* x  = (const float*)d_in[0];   // (32,64,64,64) f32 NCHW
    const float* cb = (const float*)d_in[1];   // (512,64) f32

    // d_out layout: [loss(1)][quantized_st(8388608)][perplexity(1)][encodings(67108864)]
    float* out      = (float*)d_out;
    float* out_loss = out;
    float* qout     = out + 1;
    float* out_perp = out + 1 + 8388608;
    float* enc      = out + 2 + 8388608;

    // scratch layout
    float*    hnorm    = (float*)d_ws;                  // 512 f32
    unsigned* counts   = (unsigned*)(hnorm + NCODE);    // 512 u32
    float*    partials = (float*)(counts + NCODE);      // NBLK f32

    vq_prep <<<8,    256, 0, stream>>>(cb, hnorm, counts, partials);
    vq_main <<<NBLK, 128, 0, stream>>>(x, cb, hnorm, counts, partials, qout, enc);
    vq_final<<<1,    256, 0, stream>>>(counts, partials, out_loss, out_perp);
}